// ConvTrasnpose_23038204576407
// MI455X (gfx1250) — compile-verified
//
#include <hip/hip_runtime.h>

typedef __attribute__((ext_vector_type(2))) float v2f;
typedef __attribute__((ext_vector_type(8))) float v8f;

#define C_CH 128            // C_IN == C_OUT == 128
#define TILE_M 16
#define WAVES_PER_BLOCK 8
#define BLOCK_ROWS (TILE_M * WAVES_PER_BLOCK)   // 128 rows of M per wave-set
#define TILES_PER_BLOCK 4                        // amortize W_k staging
#define BN_EPS 1e-5f

#if defined(__HIP_DEVICE_COMPILE__)
#if !__has_builtin(__builtin_amdgcn_wmma_f32_16x16x4_f32)
#error "DEVICE PASS: __builtin_amdgcn_wmma_f32_16x16x4_f32 not available"
#endif
#endif

// ---------------------------------------------------------------------------
// Kernel 1: zero the fp32 output accumulator and the BN stats scratch.
// ---------------------------------------------------------------------------
__global__ __launch_bounds__(256) void zero_kernel(float4* __restrict__ out4,
                                                   long n4,
                                                   float* __restrict__ stats) {
  long idx = (long)blockIdx.x * blockDim.x + threadIdx.x;
  long stride = (long)gridDim.x * blockDim.x;
  float4 z; z.x = 0.f; z.y = 0.f; z.z = 0.f; z.w = 0.f;
  for (long i = idx; i < n4; i += stride) out4[i] = z;
  if (blockIdx.x == 0 && threadIdx.x < 2 * C_CH) stats[threadIdx.x] = 0.f;
}

// ---------------------------------------------------------------------------
// Kernel 2: gather -> WMMA f32 GEMM -> scatter-add (atomic f32).
// grid = (ceil(M/512), K3), block = 256 (8 waves of 32).
// Each block stages W_k to LDS once (bank-swizzled), then processes 4
// 128-row tiles of the map list; each wave computes a 16x128 output tile
// with chains of V_WMMA_F32_16X16X4_F32.
// ---------------------------------------------------------------------------
__global__ __launch_bounds__(256) void spconv_gemm_kernel(
    const float* __restrict__ feats,     // [N_IN, 128]
    const float* __restrict__ W,         // [K3, 128, 128]
    const int*   __restrict__ in_maps,   // [K3, M]
    const int*   __restrict__ out_maps,  // [K3, M]
    float*       __restrict__ out,       // [N_OUT, 128] accumulator
    int M) {
  __shared__ float Wlds[C_CH * C_CH];    // 64 KB: W_k staged once per block

  const int k   = blockIdx.y;
  const int tid = threadIdx.x;

  // Stage W_k into LDS with a 16-column XOR swizzle keyed on (row>>1)&1 so
  // the two lane-halves of each wave hit disjoint LDS bank groups later.
  {
    const float4* Wg = (const float4*)(W + (size_t)k * C_CH * C_CH);
    float4* Ws = (float4*)Wlds;
    #pragma unroll
    for (int i = 0; i < (C_CH * C_CH / 4) / 256; ++i) {
      int idx = tid + i * 256;
      int r    = idx >> 5;                       // row (32 float4 per row)
      int c4   = idx & 31;                       // float4 column
      int c4s  = c4 ^ (((r >> 1) & 1) << 2);     // XOR 16 floats = 4 float4
      Ws[(r << 5) + c4s] = Wg[idx];
    }
  }
  __syncthreads();

  const int wave = tid >> 5;
  const int lane = tid & 31;
  const int half = lane >> 4;        // 0: lanes 0-15, 1: lanes 16-31
  const int l15  = lane & 15;

  for (int t = 0; t < TILES_PER_BLOCK; ++t) {
    const int base =
        (blockIdx.x * TILES_PER_BLOCK + t) * BLOCK_ROWS + wave * TILE_M;
    if (base >= M) continue;   // uniform per wave: EXEC stays all-1s inside

    // ---- Gather A tile (16 rows x 128 cols) into 32 f32 WMMA fragments. --
    // A layout (32-bit A 16x4): lane<16 row=l15 holds K = 4s+0,4s+1;
    //                           lane>=16 row=l15 holds K = 4s+2,4s+3.
    int m_row = base + l15;
    int m_cl  = m_row < M ? m_row : (M - 1);     // clamp: keep EXEC uniform
    int im    = in_maps[(size_t)k * M + m_cl];
    const float* arow = feats + (size_t)im * C_CH + 2 * half;
    v2f afrag[32];
    #pragma unroll
    for (int s = 0; s < 32; ++s)
      afrag[s] = *(const v2f*)(arow + 4 * s);

    // ---- Output rows for the scatter (D layout: VGPR v -> row v+8*half). --
    int omr[8];
    #pragma unroll
    for (int v = 0; v < 8; ++v) {
      int mr  = base + v + 8 * half;
      int mrc = mr < M ? mr : (M - 1);
      int o   = out_maps[(size_t)k * M + mrc];
      omr[v]  = (mr < M) ? o : -1;
    }

    // ---- 8 N-tiles of 16 output channels each. ----
    for (int nt = 0; nt < 8; ++nt) {
      const int col = nt * 16 + l15;
      // Swizzled read column: for rows 4s+2*half / +1, (row>>1)&1 == half.
      const int cs = col ^ (half << 4);
      v8f acc = {};
      // Two chunks of 16 K-steps: batch all B loads, then run the WMMA chain
      // back-to-back (single wait per chunk, XDL pipeline stays full).
      #pragma unroll
      for (int h = 0; h < 2; ++h) {
        v2f bfrag[16];
        #pragma unroll
        for (int s2 = 0; s2 < 16; ++s2) {
          int r0 = 4 * (h * 16 + s2) + 2 * half;
          bfrag[s2].x = Wlds[(r0 + 0) * C_CH + cs];
          bfrag[s2].y = Wlds[(r0 + 1) * C_CH + cs];
        }
        #pragma unroll
        for (int s2 = 0; s2 < 16; ++s2) {
          acc = __builtin_amdgcn_wmma_f32_16x16x4_f32(
              /*neg_a=*/false, afrag[h * 16 + s2], /*neg_b=*/false, bfrag[s2],
              /*c_mod=*/(short)0, acc, /*reuse_a=*/false, /*reuse_b=*/false);
        }
      }
      // Scatter-add 16x16 tile: lane holds column `col` for rows v (+8*half).
      #pragma unroll
      for (int v = 0; v < 8; ++v) {
        if (omr[v] >= 0)
          atomicAdd(out + (size_t)omr[v] * C_CH + col, acc[v]);
      }
    }
  }
}

// ---------------------------------------------------------------------------
// Kernel 3: per-channel sum / sum-of-squares (coalesced: thread == channel).
// ---------------------------------------------------------------------------
__global__ __launch_bounds__(C_CH) void bn_stats_kernel(
    const float* __restrict__ out, float* __restrict__ stats, int n_rows) {
  const int c = threadIdx.x;
  float s = 0.f, sq = 0.f;
  for (int r = blockIdx.x; r < n_rows; r += gridDim.x) {
    float v = out[(size_t)r * C_CH + c];
    s += v;
    sq += v * v;
  }
  atomicAdd(&stats[c], s);
  atomicAdd(&stats[C_CH + c], sq);
}

// ---------------------------------------------------------------------------
// Kernel 4: fold stats + gamma/beta into fused scale/bias per channel.
// ---------------------------------------------------------------------------
__global__ __launch_bounds__(C_CH) void bn_finalize_kernel(
    const float* __restrict__ stats, const float* __restrict__ gamma,
    const float* __restrict__ beta, float* __restrict__ sb, int n_rows) {
  const int c = threadIdx.x;
  float inv_n = 1.0f / (float)n_rows;
  float mean  = stats[c] * inv_n;
  float var   = stats[C_CH + c] * inv_n - mean * mean;
  float scale = gamma[c] * rsqrtf(var + BN_EPS);
  sb[c]        = scale;
  sb[C_CH + c] = beta[c] - mean * scale;
}

// ---------------------------------------------------------------------------
// Kernel 5: y = x * scale[c] + bias[c], in place, float4 vectorized.
// ---------------------------------------------------------------------------
__global__ __launch_bounds__(256) void bn_apply_kernel(
    float4* __restrict__ out4, const float4* __restrict__ sb4, long n4) {
  long idx = (long)blockIdx.x * blockDim.x + threadIdx.x;
  long stride = (long)gridDim.x * blockDim.x;
  for (long i = idx; i < n4; i += stride) {
    int c4 = (int)(i & (C_CH / 4 - 1));   // 32 float4 groups per row
    float4 v = out4[i];
    float4 sc = sb4[c4];
    float4 bs = sb4[C_CH / 4 + c4];
    v.x = v.x * sc.x + bs.x;
    v.y = v.y * sc.y + bs.y;
    v.z = v.z * sc.z + bs.z;
    v.w = v.w * sc.w + bs.w;
    out4[i] = v;
  }
}

// ---------------------------------------------------------------------------
extern "C" void kernel_launch(void* const* d_in, const int* in_sizes, int n_in,
                              void* d_out, int out_size, void* d_ws, size_t ws_size,
                              hipStream_t stream) {
  const float* feats    = (const float*)d_in[0];
  const float* W        = (const float*)d_in[1];
  const float* gamma    = (const float*)d_in[2];
  const float* beta     = (const float*)d_in[3];
  const int*   in_maps  = (const int*)d_in[4];
  const int*   out_maps = (const int*)d_in[5];
  // d_in[6] is n_out on device; derive it from out_size instead (no sync copies).

  const int k3     = in_sizes[1] / (C_CH * C_CH);   // 27
  const int M      = in_sizes[4] / k3;              // 100000
  const int n_rows = out_size / C_CH;               // 400000

  float* out   = (float*)d_out;
  float* stats = (float*)d_ws;                      // [2*C_CH] sum / sumsq
  float* sb    = stats + 2 * C_CH;                  // [2*C_CH] scale / bias

  const long n4 = (long)out_size / 4;

  zero_kernel<<<4096, 256, 0, stream>>>((float4*)out, n4, stats);

  const int rows_per_block = BLOCK_ROWS * TILES_PER_BLOCK;   // 512
  dim3 gemm_grid((M + rows_per_block - 1) / rows_per_block, k3);
  spconv_gemm_kernel<<<gemm_grid, 256, 0, stream>>>(feats, W, in_maps,
                                                    out_maps, out, M);

  bn_stats_kernel<<<1024, C_CH, 0, stream>>>(out, stats, n_rows);
  bn_finalize_kernel<<<1, C_CH, 0, stream>>>(stats, gamma, beta, sb, n_rows);
  bn_apply_kernel<<<4096, 256, 0, stream>>>((float4*)out, (const float4*)sb, n4);
}